// CDVectorQuantizer_12945031430911
// MI455X (gfx1250) — compile-verified
//
#include <hip/hip_runtime.h>

// CDNA5 / gfx1250. fp32 VQ distance argmin via V_WMMA_F32_16X16X4_F32.
// 4 independent WMMA accumulator chains per wave + global_prefetch of the
// next codebook stage to overlap LDS staging with matrix compute.

typedef __attribute__((ext_vector_type(2))) float v2f;
typedef __attribute__((ext_vector_type(8))) float v8f;

#define B_      32
#define D_      256
#define T_      4096
#define K_      1024
#define TT      128        // time positions per workgroup
#define CT      64         // codebook rows staged per LDS stage
#define STRIDE  260        // padded LDS row stride in floats (bank-conflict-free)
#define NSTAGE  (K_ / CT)  // 16 codebook stages

// ---------------------------------------------------------------------------
// Kernel 0: per-code squared norms  ||e_c||^2  -> d_ws (1024 floats)
// ---------------------------------------------------------------------------
__global__ __launch_bounds__(1024) void vq_norms_kernel(const float* __restrict__ emb,
                                                        float* __restrict__ norms) {
    int r = blockIdx.x * blockDim.x + threadIdx.x;
    if (r < K_) {
        const float4* row = (const float4*)(emb + (size_t)r * D_);
        float s = 0.f;
#pragma unroll 8
        for (int i = 0; i < D_ / 4; ++i) {
            float4 v = row[i];
            s += v.x * v.x + v.y * v.y + v.z * v.z + v.w * v.w;
        }
        norms[r] = s;
    }
}

// ---------------------------------------------------------------------------
// Kernel 1: fused distance-matmul (WMMA fp32) + argmin + gather + transpose
// Grid: (B_*T_)/TT workgroups of 256 threads (8 waves, wave32).
// Each wave computes 4 concurrent 16(codes) x 16(times) WMMA score tiles.
// ---------------------------------------------------------------------------
__global__ __launch_bounds__(256) void vq_argmin_kernel(const float* __restrict__ z,
                                                        const float* __restrict__ emb,
                                                        const float* __restrict__ enorm,
                                                        float* __restrict__ out) {
    extern __shared__ float smem[];
    float* zs   = smem;                    // [TT][STRIDE]  z tile, [t][d]
    float* es   = zs + TT * STRIDE;        // [CT][STRIDE]  codebook tile, [c][d]
    float* esn  = es + CT * STRIDE;        // [CT]          staged ||e||^2
    int*   sidx = (int*)(esn + CT);        // [TT]          per-time argmin

    const int tid    = threadIdx.x;
    const int wave   = tid >> 5;
    const int lane   = tid & 31;
    const int laneLo = lane & 15;
    const int hi     = lane >> 4;          // 0: lanes 0-15, 1: lanes 16-31

    const int wg = blockIdx.x;
    const int b  = wg >> 5;                // T_/TT == 32 tiles per batch
    const int t0 = (wg & 31) * TT;

    // ---- stage z tile: zs[t][d] = z[b, d, t0+t]  (coalesced 512B row chunks)
    const float* zb = z + (size_t)b * D_ * T_ + t0;
    for (int i = tid; i < TT * D_; i += 256) {
        int d = i >> 7;            // i / TT
        int t = i & (TT - 1);
        zs[t * STRIDE + d] = zb[(size_t)d * T_ + t];
    }

    float minv = 3.4e38f;
    int   mini = 0;

    const int tRow = wave * 16 + laneLo;   // this wave's local time row

    for (int stage = 0; stage < NSTAGE; ++stage) {
        __syncthreads();                   // prev-stage reads done / z tile visible
        // ---- stage codebook tile: es[c][d] = emb[(stage*CT+c)*D_ + d]
        const float* eb = emb + (size_t)stage * CT * D_;
        for (int i = tid; i < CT * D_; i += 256) {
            int c = i >> 8;        // i / D_
            int d = i & (D_ - 1);
            es[c * STRIDE + d] = eb[i];
        }
        if (tid < CT) esn[tid] = enorm[stage * CT + tid];
        __syncthreads();

        // ---- warm L2/L1 with next stage's 64KB while WMMAs run
        if (stage + 1 < NSTAGE) {
            const float* nxt = emb + (size_t)(stage + 1) * CT * D_ + (size_t)tid * 64;
            __builtin_prefetch(nxt, 0, 1);        // global_prefetch_b8
            __builtin_prefetch(nxt + 32, 0, 1);   // second 128B line of 256B chunk
        }

        // ---- 4 code subtiles of 16 as 4 independent WMMA chains (shared B)
        v8f acc0 = {}, acc1 = {}, acc2 = {}, acc3 = {};
        // A layout per step s (d0=4s): VGPR0 = {d0 | d0+2}, VGPR1 = {d0+1 | d0+3}
        const float* aRow = es + laneLo * STRIDE + 2 * hi;
        // B layout identical in K: one ds_load_b64 per step, reused 4x
        const float* bRow = zs + tRow * STRIDE + 2 * hi;
#pragma unroll 8
        for (int s = 0; s < D_ / 4; ++s) {
            v2f vb  = *(const v2f*)(bRow + 4 * s);
            v2f va0 = *(const v2f*)(aRow + 4 * s);
            v2f va1 = *(const v2f*)(aRow + 16 * STRIDE + 4 * s);
            v2f va2 = *(const v2f*)(aRow + 32 * STRIDE + 4 * s);
            v2f va3 = *(const v2f*)(aRow + 48 * STRIDE + 4 * s);
            acc0 = __builtin_amdgcn_wmma_f32_16x16x4_f32(
                       false, va0, false, vb, (short)0, acc0, false, false);
            acc1 = __builtin_amdgcn_wmma_f32_16x16x4_f32(
                       false, va1, false, vb, (short)0, acc1, false, false);
            acc2 = __builtin_amdgcn_wmma_f32_16x16x4_f32(
                       false, va2, false, vb, (short)0, acc2, false, false);
            acc3 = __builtin_amdgcn_wmma_f32_16x16x4_f32(
                       false, va3, false, vb, (short)0, acc3, false, false);
        }

        // ---- argmin update, codes ascending (ties keep the first/lowest index)
        // C/D layout: VGPR v holds code (v | v+8) across lane halves, time=laneLo
        const int cbase = stage * CT + 8 * hi;
#pragma unroll
        for (int v = 0; v < 8; ++v) {
            float dv = esn[8 * hi + v] - 2.0f * acc0[v];
            if (dv < minv) { minv = dv; mini = cbase + v; }
        }
#pragma unroll
        for (int v = 0; v < 8; ++v) {
            float dv = esn[16 + 8 * hi + v] - 2.0f * acc1[v];
            if (dv < minv) { minv = dv; mini = cbase + 16 + v; }
        }
#pragma unroll
        for (int v = 0; v < 8; ++v) {
            float dv = esn[32 + 8 * hi + v] - 2.0f * acc2[v];
            if (dv < minv) { minv = dv; mini = cbase + 32 + v; }
        }
#pragma unroll
        for (int v = 0; v < 8; ++v) {
            float dv = esn[48 + 8 * hi + v] - 2.0f * acc3[v];
            if (dv < minv) { minv = dv; mini = cbase + 48 + v; }
        }
    }

    // ---- merge lane halves (codes v vs v+8), tie-break toward lower index
    float ov = __shfl_xor(minv, 16, 32);
    int   oi = __shfl_xor(mini, 16, 32);
    if (ov < minv || (ov == minv && oi < mini)) { minv = ov; mini = oi; }

    __syncthreads();
    if (lane < 16) sidx[wave * 16 + laneLo] = mini;
    __syncthreads();

    // ---- gather codebook rows and write transposed: out[b, d, t0+t]
    float* ob = out + (size_t)b * D_ * T_ + t0;
    for (int i = tid; i < TT * D_; i += 256) {
        int d = i >> 7;
        int t = i & (TT - 1);
        ob[(size_t)d * T_ + t] = emb[(size_t)sidx[t] * D_ + d];
    }
}

// ---------------------------------------------------------------------------
extern "C" void kernel_launch(void* const* d_in, const int* in_sizes, int n_in,
                              void* d_out, int out_size, void* d_ws, size_t ws_size,
                              hipStream_t stream) {
    const float* z   = (const float*)d_in[0];   // [32, 256, 4096] fp32
    const float* emb = (const float*)d_in[1];   // [1024, 256]     fp32
    float*       out = (float*)d_out;           // [32, 256, 4096] fp32
    float*       nrm = (float*)d_ws;            // 1024 floats of scratch

    vq_norms_kernel<<<dim3(1), dim3(1024), 0, stream>>>(emb, nrm);

    size_t smem = (size_t)(TT * STRIDE + CT * STRIDE + CT) * sizeof(float)
                + (size_t)TT * sizeof(int);     // ~200 KB of the 320 KB WGP LDS
    vq_argmin_kernel<<<dim3((B_ * T_) / TT), dim3(256), smem, stream>>>(z, emb, nrm, out);
}